// GRNGCN_81381040325515
// MI455X (gfx1250) — compile-verified
//
#include <hip/hip_runtime.h>

typedef __attribute__((ext_vector_type(2))) float v2f;
typedef __attribute__((ext_vector_type(8))) float v8f;

#define CH_IN 128

// ---------------- degree / normalization ----------------

__global__ __launch_bounds__(256) void grngcn_deg_init(float* __restrict__ deg, int n) {
    int i = blockIdx.x * blockDim.x + threadIdx.x;
    if (i < n) deg[i] = 1.0f;  // self-loop
}

__global__ __launch_bounds__(256) void grngcn_deg_count(float* __restrict__ deg,
                                                        const int* __restrict__ dst, int E) {
    int i = blockIdx.x * blockDim.x + threadIdx.x;
    if (i < E) atomicAdd(&deg[dst[i]], 1.0f);
}

__global__ __launch_bounds__(256) void grngcn_rsqrt(float* __restrict__ dinv, int n) {
    int i = blockIdx.x * blockDim.x + threadIdx.x;
    if (i < n) dinv[i] = rsqrtf(dinv[i]);
}

// ---------------- fused GEMM (+optional ReLU on A) + bias + self-loop init ----------------
// One wave computes one 16x16 tile of H = A(+relu) @ W, K = 128, via
// V_WMMA_F32_16X16X4_F32 (full fp32 precision). Stores H and the scatter
// accumulator Z pre-seeded with H*dinv^2 + bias.
//
// fp32 WMMA VGPR layouts (ISA 7.12.2):
//   A 16x4 : lanes 0-15 -> K = {k, k+1}, lanes 16-31 -> K = {k+2, k+3}; M = lane%16
//   B 4x16 : same K split across lane halves; N = lane%16
//   C/D    : VGPR i -> row m0 + i + 8*(lane/16), col n0 + lane%16

template <int NOUT, bool RELU>
__global__ __launch_bounds__(NOUT * 2) void grngcn_gemm(
    const float* __restrict__ A, const float* __restrict__ W,
    const float* __restrict__ bias, const float* __restrict__ dinv,
    float* __restrict__ H, float* __restrict__ Z, int nrows) {
    const int K    = CH_IN;
    const int lane = threadIdx.x & 31;
    const int wave = threadIdx.x >> 5;
    const int m0   = blockIdx.x * 16;
    const int n0   = wave * 16;
    const int half = lane >> 4;
    const int r    = lane & 15;

    int arow = m0 + r;
    if (arow >= nrows) arow = nrows - 1;  // keep EXEC all-ones for WMMA
    const float* Ap = A + (size_t)arow * K;

    v8f c = {};
#pragma unroll 4
    for (int k = 0; k < K; k += 4) {
        const int kk = k + 2 * half;
        float a0 = Ap[kk];
        float a1 = Ap[kk + 1];
        if (RELU) { a0 = fmaxf(a0, 0.0f); a1 = fmaxf(a1, 0.0f); }
        v2f a, b;
        a.x = a0;
        a.y = a1;
        b.x = W[(size_t)kk * NOUT + n0 + r];
        b.y = W[(size_t)(kk + 1) * NOUT + n0 + r];
        c = __builtin_amdgcn_wmma_f32_16x16x4_f32(false, a, false, b, (short)0, c,
                                                  false, false);
    }

    const float bn = bias[n0 + r];
#pragma unroll
    for (int i = 0; i < 8; ++i) {
        const int m = m0 + i + 8 * half;
        if (m < nrows) {
            const float di = dinv[m];
            const float v  = c[i];
            H[(size_t)m * NOUT + n0 + r] = v;
            Z[(size_t)m * NOUT + n0 + r] = v * di * di + bn;
        }
    }
}

// ---------------- edge scatter: Z[dst] += H[src] * dinv[src]*dinv[dst] ----------------
// One wave per edge; each lane owns C/32 contiguous channels (vectorized gather,
// fp32 global atomics for the scatter-add).

template <int C>
__global__ __launch_bounds__(256) void grngcn_scatter(
    const float* __restrict__ H, float* __restrict__ Z,
    const int* __restrict__ src, const int* __restrict__ dst,
    const float* __restrict__ dinv, int E) {
    const int gw   = (int)((blockIdx.x * blockDim.x + threadIdx.x) >> 5);
    const int lane = threadIdx.x & 31;
    if (gw >= E) return;
    const int s = src[gw];
    const int d = dst[gw];
    const float norm = dinv[s] * dinv[d];
    constexpr int PER = C / 32;  // 4 (128ch) or 2 (64ch)
    const float* hs = H + (size_t)s * C + lane * PER;
    float* zd       = Z + (size_t)d * C + lane * PER;
    float v[PER];
#pragma unroll
    for (int i = 0; i < PER; ++i) v[i] = hs[i];
#pragma unroll
    for (int i = 0; i < PER; ++i) atomicAdd(&zd[i], v[i] * norm);
}

// ---------------- decoder: out[e] = dot(z2[src_e], z2[dst_e]) over 64 ch ----------------
// One wave per edge; float2 per lane, wave32 xor-butterfly reduction.

__global__ __launch_bounds__(256) void grngcn_decode(
    const float* __restrict__ Z, const int* __restrict__ pos,
    const int* __restrict__ neg, int E, float* __restrict__ out) {
    const int gw   = (int)((blockIdx.x * blockDim.x + threadIdx.x) >> 5);
    const int lane = threadIdx.x & 31;
    if (gw >= 2 * E) return;
    int s, d;
    if (gw < E) {
        s = pos[gw];
        d = pos[gw + E];
    } else {
        const int e = gw - E;
        s = neg[e];
        d = neg[e + E];
    }
    const float* zs = Z + (size_t)s * 64 + lane * 2;
    const float* zd = Z + (size_t)d * 64 + lane * 2;
    float acc = zs[0] * zd[0] + zs[1] * zd[1];
#pragma unroll
    for (int off = 16; off > 0; off >>= 1) acc += __shfl_xor(acc, off, 32);
    if (lane == 0) out[gw] = acc;
}

// ---------------- launcher ----------------

extern "C" void kernel_launch(void* const* d_in, const int* in_sizes, int n_in,
                              void* d_out, int out_size, void* d_ws, size_t ws_size,
                              hipStream_t stream) {
    const float* x  = (const float*)d_in[0];
    const int*   pos = (const int*)d_in[1];   // [2, E] row-major: src then dst
    const int*   neg = (const int*)d_in[2];
    const float* W1 = (const float*)d_in[3];  // [128,128] row-major
    const float* b1 = (const float*)d_in[4];  // [128]
    const float* W2 = (const float*)d_in[5];  // [128,64]
    const float* b2 = (const float*)d_in[6];  // [64]

    const int N = in_sizes[0] / CH_IN;  // 100000 nodes
    const int E = in_sizes[1] / 2;      // 600000 pos edges

    // workspace layout (floats): dinv[N] | bufA[N*128] | bufB[N*128]
    float* dinv = (float*)d_ws;
    float* bufA = dinv + N;                    // h1; later h2 (N*64) + z2 (N*64)
    float* bufB = bufA + (size_t)N * CH_IN;    // z1 accumulator
    float* h2   = bufA;
    float* z2   = bufA + (size_t)N * 64;

    // 1) degrees -> dinv = deg^-1/2 (self-loop included)
    grngcn_deg_init<<<(N + 255) / 256, 256, 0, stream>>>(dinv, N);
    grngcn_deg_count<<<(E + 255) / 256, 256, 0, stream>>>(dinv, pos + E, E);
    grngcn_rsqrt<<<(N + 255) / 256, 256, 0, stream>>>(dinv, N);

    // 2) layer 1: h1 = x@W1 ; z1 = h1*dinv^2 + b1 ; then scatter edge messages
    grngcn_gemm<128, false><<<(N + 15) / 16, 256, 0, stream>>>(x, W1, b1, dinv,
                                                               bufA, bufB, N);
    grngcn_scatter<128><<<(E + 7) / 8, 256, 0, stream>>>(bufA, bufB, pos, pos + E,
                                                         dinv, E);

    // 3) layer 2 (ReLU fused into A-load): h2 = relu(z1)@W2 ; z2 seeded ; scatter
    grngcn_gemm<64, true><<<(N + 15) / 16, 128, 0, stream>>>(bufB, W2, b2, dinv,
                                                             h2, z2, N);
    grngcn_scatter<64><<<(E + 7) / 8, 256, 0, stream>>>(h2, z2, pos, pos + E,
                                                        dinv, E);

    // 4) decode all pos+neg edges
    grngcn_decode<<<((2 * E) + 7) / 8, 256, 0, stream>>>(z2, pos, neg, E,
                                                         (float*)d_out);
}